// MHSelfAttention_17239998726666
// MI455X (gfx1250) — compile-verified
//
#include <hip/hip_runtime.h>
#include <math.h>

// ---- CDNA5 (gfx1250) wave32 WMMA fp32 path -------------------------------
typedef __attribute__((ext_vector_type(2))) float v2f;
typedef __attribute__((ext_vector_type(4))) float v4f;
typedef __attribute__((ext_vector_type(8))) float v8f;

static constexpr int Bn  = 4;
static constexpr int Sn  = 1024;
static constexpr int Dn  = 1024;
static constexpr int Hn  = 16;
static constexpr int HDn = 64;
static constexpr float QK_SCALE = 0.125f;  // 1/sqrt(64)

__device__ __forceinline__ v8f wmma4(v2f a, v2f b, v8f c) {
  // V_WMMA_F32_16X16X4_F32 : D = A(16x4,f32) x B(4x16,f32) + C(16x16,f32)
  return __builtin_amdgcn_wmma_f32_16x16x4_f32(
      /*neg_a=*/false, a, /*neg_b=*/false, b,
      /*c_mod=*/(short)0, c, /*reuse_a=*/false, /*reuse_b=*/false);
}
__device__ __forceinline__ v8f zero8() {
  v8f z = {0.f, 0.f, 0.f, 0.f, 0.f, 0.f, 0.f, 0.f};
  return z;
}

// Async copy of 16 bytes (global -> LDS), tracked by ASYNCcnt.
__device__ __forceinline__ void async_copy_b128(const float* gsrc, float* ldst) {
  const unsigned lds = (unsigned)(uintptr_t)ldst;          // low 32 bits give the LDS address
  const unsigned long long ga = (unsigned long long)(uintptr_t)gsrc;
  asm volatile("global_load_async_to_lds_b128 %0, %1, off" ::"v"(lds), "v"(ga)
               : "memory");
}
__device__ __forceinline__ void wait_async0() {
  asm volatile("s_wait_asynccnt 0x0" ::: "memory");
}

// ---------------------------------------------------------------------------
// Kernel 1: fused per-head QKV projection.
// grid = (S/128, B*H), block = 256 (8 waves). Wave w owns rows [bx*128+16w,+16).
// Weight K-slabs (3 x 64x64 fp32 = 48 KB) are double-buffered in LDS via
// global_load_async_to_lds_b128; copy of tile t+1 overlaps WMMAs of tile t.
// Scale 1/sqrt(hd) folded into Q.
// ---------------------------------------------------------------------------
__global__ __launch_bounds__(256) void mhsa_qkv(
    const float* __restrict__ x,
    const float* __restrict__ wq, const float* __restrict__ biasq,
    const float* __restrict__ wk, const float* __restrict__ biask,
    const float* __restrict__ wv, const float* __restrict__ biasv,
    float* __restrict__ Qw, float* __restrict__ Kw, float* __restrict__ Vw) {
  __shared__ __align__(16) float wsm[2][3][64 * HDn];  // 2 x 3 x 16 KB = 96 KB

  const int t    = threadIdx.x;
  const int lane = t & 31;
  const int wave = t >> 5;
  const int l15  = lane & 15;
  const int hi   = lane >> 4;                 // 0: K pair {0,1}, 1: K pair {2,3}
  const int bh   = blockIdx.y;                // b*H + h
  const int h    = bh & (Hn - 1);
  const int b    = bh >> 4;
  const int m0   = blockIdx.x * 128 + wave * 16;

  // Stage one 64-deep K-slab of wq/wk/wv into LDS buffer `buf` (async).
  auto stage = [&](int buf, int kt) {
    const size_t base = ((size_t)h * Dn + kt * 64) * HDn;  // contiguous 16 KB
    const float* srcs[3] = {wq + base, wk + base, wv + base};
#pragma unroll
    for (int mIdx = 0; mIdx < 3; ++mIdx) {
#pragma unroll
      for (int it = 0; it < 4; ++it) {
        const int off = it * 1024 + t * 4;    // 4096 floats per matrix
        async_copy_b128(srcs[mIdx] + off, &wsm[buf][mIdx][off]);
      }
    }
  };

  v8f qa[4], ka[4], va[4];
#pragma unroll
  for (int n = 0; n < 4; ++n) { qa[n] = zero8(); ka[n] = zero8(); va[n] = zero8(); }

  const float* xrow = x + ((size_t)b * Sn + (m0 + l15)) * Dn + 2 * hi;

  stage(0, 0);
  for (int kt = 0; kt < Dn / 64; ++kt) {
    wait_async0();        // this wave's async copies of tile kt have landed
    __syncthreads();      // ... and every other wave's, too
    if (kt + 1 < Dn / 64) stage((kt + 1) & 1, kt + 1);
    const int buf = kt & 1;

    for (int k0 = kt * 64; k0 < kt * 64 + 64; k0 += 4) {
      if ((k0 & 127) == 0)
        __builtin_prefetch(xrow + k0 + 128, 0, 3);  // global_prefetch_b8 (near)
      const v2f a = *reinterpret_cast<const v2f*>(xrow + k0);
      const int kk = (k0 - kt * 64) + 2 * hi;       // local K row in slab
#pragma unroll
      for (int n = 0; n < 4; ++n) {
        const int col = n * 16 + l15;
        const float* pq = &wsm[buf][0][kk * HDn + col];
        const float* pk = &wsm[buf][1][kk * HDn + col];
        const float* pv = &wsm[buf][2][kk * HDn + col];
        v2f bq; bq.x = pq[0]; bq.y = pq[HDn];
        v2f bk; bk.x = pk[0]; bk.y = pk[HDn];
        v2f bv; bv.x = pv[0]; bv.y = pv[HDn];
        qa[n] = wmma4(a, bq, qa[n]);
        ka[n] = wmma4(a, bk, ka[n]);
        va[n] = wmma4(a, bv, va[n]);
      }
    }
  }

#pragma unroll
  for (int n = 0; n < 4; ++n) {
    const int col = n * 16 + l15;
    const float cq = biasq[h * HDn + col];
    const float ck = biask[h * HDn + col];
    const float cv = biasv[h * HDn + col];
    float* oq = Qw + ((size_t)bh * Sn + m0 + 8 * hi) * HDn + col;
    float* ok = Kw + ((size_t)bh * Sn + m0 + 8 * hi) * HDn + col;
    float* ov = Vw + ((size_t)bh * Sn + m0 + 8 * hi) * HDn + col;
#pragma unroll
    for (int r = 0; r < 8; ++r) {                // D-matrix: M = r + 8*hi
      oq[(size_t)r * HDn] = (qa[n][r] + cq) * QK_SCALE;
      ok[(size_t)r * HDn] = ka[n][r] + ck;
      ov[(size_t)r * HDn] = va[n][r] + cv;
    }
  }
}

// ---------------------------------------------------------------------------
// Kernel 2: per (b,h) 16-row block: scores -> LDS -> softmax stats -> att*V.
// grid = (S/16, B*H), block = 128 (4 waves). 64 KB LDS score tile.
// Writes out_h and the (m, l) stats reused by kernel 3.
// ---------------------------------------------------------------------------
__global__ __launch_bounds__(128) void mhsa_attn(
    const float* __restrict__ Qw, const float* __restrict__ Kw,
    const float* __restrict__ Vw,
    float* __restrict__ Mw, float* __restrict__ Lw,
    float* __restrict__ OH) {
  __shared__ __align__(16) float sc[16 * Sn];   // 16 rows x 1024 scores (64 KB)

  const int lane = threadIdx.x & 31;
  const int wave = threadIdx.x >> 5;            // 0..3
  const int l15  = lane & 15;
  const int hi   = lane >> 4;
  const int bh   = blockIdx.y;
  const int i0   = blockIdx.x * 16;

  // Preload this block's Q tile (16 x 64) into registers: 16 K-chunks of v2f.
  v2f qreg[16];
  const float* qrow = Qw + ((size_t)bh * Sn + i0 + l15) * HDn + 2 * hi;
#pragma unroll
  for (int c = 0; c < 16; ++c) qreg[c] = *reinterpret_cast<const v2f*>(qrow + 4 * c);

  // Scores: wave w handles j-blocks w, w+4, ... (16 of 64).
  for (int jb = wave; jb < Sn / 16; jb += 4) {
    const int j0 = jb * 16;
    v8f s = zero8();
    const float* krow = Kw + ((size_t)bh * Sn + j0 + l15) * HDn + 2 * hi;
#pragma unroll
    for (int c = 0; c < 16; ++c) {
      const v2f bm = *reinterpret_cast<const v2f*>(krow + 4 * c);
      s = wmma4(qreg[c], bm, s);
    }
    float* dst = sc + (size_t)(8 * hi) * Sn + j0 + l15;
#pragma unroll
    for (int r = 0; r < 8; ++r) dst[(size_t)r * Sn] = s[r];
  }
  __syncthreads();

  // Softmax stats: 8 lanes per row (128 threads / 16 rows), shuffle reduce.
  {
    const int row = threadIdx.x >> 3;
    const int seg = threadIdx.x & 7;
    float* rsc = sc + (size_t)row * Sn;
    float mx = -3.4e38f;
    for (int j = seg * 128; j < seg * 128 + 128; ++j) mx = fmaxf(mx, rsc[j]);
#pragma unroll
    for (int d = 1; d < 8; d <<= 1) mx = fmaxf(mx, __shfl_xor(mx, d, 32));
    float sum = 0.f;
    for (int j = seg * 128; j < seg * 128 + 128; ++j) {
      const float e = __expf(rsc[j] - mx);
      rsc[j] = e;
      sum += e;
    }
#pragma unroll
    for (int d = 1; d < 8; d <<= 1) sum += __shfl_xor(sum, d, 32);
    if (seg == 0) {
      Mw[(size_t)bh * Sn + i0 + row] = mx;
      Lw[(size_t)bh * Sn + i0 + row] = sum;
    }
  }
  __syncthreads();

  // att * V : wave w owns column block w (16 of 64 head dims).
  const int col = wave * 16 + l15;
  v8f acc = zero8();
  for (int j0 = 0; j0 < Sn; j0 += 4) {
    const int jj = j0 + 2 * hi;
    const v2f a = *reinterpret_cast<const v2f*>(sc + (size_t)l15 * Sn + jj);
    const float* vp = Vw + ((size_t)bh * Sn + jj) * HDn + col;
    v2f bm; bm.x = vp[0]; bm.y = vp[HDn];
    acc = wmma4(a, bm, acc);
  }
  const float* lp = Lw + (size_t)bh * Sn + i0 + 8 * hi;
  float* op = OH + ((size_t)bh * Sn + i0 + 8 * hi) * HDn + col;
#pragma unroll
  for (int r = 0; r < 8; ++r) op[(size_t)r * HDn] = acc[r] / lp[r];
}

// ---------------------------------------------------------------------------
// Kernel 3: recompute 16x16 score tiles for ALL 16 heads, normalize with
// (m,l), stage [i][j][h] in LDS, write [B,S,S,H] with coalesced non-temporal
// b128 bursts (268 MB stream; keep it out of L2).
// grid = (S/16 j, S/16 i, B), block = 256 (8 waves, 2 heads each).
// ---------------------------------------------------------------------------
__global__ __launch_bounds__(256) void mhsa_attw(
    const float* __restrict__ Qw, const float* __restrict__ Kw,
    const float* __restrict__ Mw, const float* __restrict__ Lw,
    float* __restrict__ att) {
  __shared__ __align__(16) float tile[16 * 16 * 16];  // [i][j][h], 16 KB
  __shared__ float m_s[256];                          // [h][i]
  __shared__ float l_s[256];

  const int t    = threadIdx.x;
  const int lane = t & 31;
  const int wave = t >> 5;
  const int l15  = lane & 15;
  const int hi   = lane >> 4;
  const int b    = blockIdx.z;
  const int i0   = blockIdx.y * 16;
  const int j0   = blockIdx.x * 16;

  {
    const int h = t >> 4, r = t & 15;
    const size_t idx = (size_t)(b * Hn + h) * Sn + i0 + r;
    m_s[t] = Mw[idx];
    l_s[t] = Lw[idx];
  }
  __syncthreads();

#pragma unroll
  for (int hh = 0; hh < 2; ++hh) {
    const int h  = wave * 2 + hh;
    const int bh = b * Hn + h;
    v8f s = zero8();
    const float* qrow = Qw + ((size_t)bh * Sn + i0 + l15) * HDn + 2 * hi;
    const float* krow = Kw + ((size_t)bh * Sn + j0 + l15) * HDn + 2 * hi;
#pragma unroll
    for (int c = 0; c < 16; ++c) {
      const v2f a  = *reinterpret_cast<const v2f*>(qrow + 4 * c);
      const v2f bm = *reinterpret_cast<const v2f*>(krow + 4 * c);
      s = wmma4(a, bm, s);
    }
#pragma unroll
    for (int r = 0; r < 8; ++r) {
      const int irow = r + 8 * hi;
      const float m    = m_s[h * 16 + irow];
      const float linv = 1.0f / l_s[h * 16 + irow];
      tile[(irow * 16 + l15) * 16 + h] = __expf(s[r] - m) * linv;
    }
  }
  __syncthreads();

  // 256 threads x 16 contiguous floats: [b, i0+i, j0+(t&15), h=0..15]
  const int i   = t >> 4;
  const int c16 = (t & 15) * 16;
  float* dst = att + ((size_t)(b * Sn + i0 + i) * Sn + j0) * Hn + c16;
  const float* src = tile + i * 256 + c16;
#pragma unroll
  for (int q = 0; q < 16; q += 4) {
    const v4f val = *reinterpret_cast<const v4f*>(src + q);
    __builtin_nontemporal_store(val, reinterpret_cast<v4f*>(dst + q));
  }
}

// ---------------------------------------------------------------------------
// Kernel 4: output projection out = concat(out_h) @ ow_w^T + ow_b.
// grid = (D/64, B*S/128), block = 256 (8 waves x 16 rows x 64 cols).
// ---------------------------------------------------------------------------
__global__ __launch_bounds__(256) void mhsa_oproj(
    const float* __restrict__ OH, const float* __restrict__ W,
    const float* __restrict__ bias, float* __restrict__ out) {
  const int lane = threadIdx.x & 31;
  const int wave = threadIdx.x >> 5;
  const int l15  = lane & 15;
  const int hi   = lane >> 4;
  const int e0   = blockIdx.x * 64;
  const int row0 = blockIdx.y * 128 + wave * 16;

  const int sg = row0 + l15;            // global row in [0, B*S)
  const int b  = sg >> 10;
  const int s  = sg & (Sn - 1);

  v8f acc[4];
#pragma unroll
  for (int n = 0; n < 4; ++n) acc[n] = zero8();

  for (int k0 = 0; k0 < Dn; k0 += 4) {
    const int d  = k0 + 2 * hi;
    const int h  = d >> 6;
    const int dd = d & (HDn - 1);       // pair never crosses a head boundary
    const v2f a = *reinterpret_cast<const v2f*>(
        OH + ((size_t)(b * Hn + h) * Sn + s) * HDn + dd);
#pragma unroll
    for (int n = 0; n < 4; ++n) {
      const int e = e0 + n * 16 + l15;
      const v2f bm = *reinterpret_cast<const v2f*>(W + (size_t)e * Dn + d);
      acc[n] = wmma4(a, bm, acc[n]);
    }
  }

#pragma unroll
  for (int n = 0; n < 4; ++n) {
    const int e = e0 + n * 16 + l15;
    const float bb = bias[e];
    float* dst = out + (size_t)(row0 + 8 * hi) * Dn + e;
#pragma unroll
    for (int r = 0; r < 8; ++r) dst[(size_t)r * Dn] = acc[n][r] + bb;
  }
}

// ---------------------------------------------------------------------------
extern "C" void kernel_launch(void* const* d_in, const int* in_sizes, int n_in,
                              void* d_out, int out_size, void* d_ws, size_t ws_size,
                              hipStream_t stream) {
  (void)in_sizes; (void)n_in; (void)out_size; (void)ws_size;

  const float* x    = (const float*)d_in[0];
  const float* wq   = (const float*)d_in[1];
  const float* bq   = (const float*)d_in[2];
  const float* wk   = (const float*)d_in[3];
  const float* bk   = (const float*)d_in[4];
  const float* wv   = (const float*)d_in[5];
  const float* bv   = (const float*)d_in[6];
  const float* ow_w = (const float*)d_in[7];
  const float* ow_b = (const float*)d_in[8];

  float* att = (float*)d_out;                              // [B,S,S,H]
  float* out = att + (size_t)Bn * Sn * Sn * Hn;            // [B,S,D]

  const size_t QKV = (size_t)Bn * Hn * Sn * HDn;           // 4 Mi floats
  float* ws = (float*)d_ws;
  float* Qw = ws;
  float* Kw = Qw + QKV;
  float* Vw = Kw + QKV;
  float* OH = Vw + QKV;
  float* Mw = OH + QKV;
  float* Lw = Mw + (size_t)Bn * Hn * Sn;

  dim3 g1(Sn / 128, Bn * Hn);
  mhsa_qkv<<<g1, 256, 0, stream>>>(x, wq, bq, wk, bk, wv, bv, Qw, Kw, Vw);

  dim3 g2(Sn / 16, Bn * Hn);
  mhsa_attn<<<g2, 128, 0, stream>>>(Qw, Kw, Vw, Mw, Lw, OH);

  dim3 g3(Sn / 16, Sn / 16, Bn);
  mhsa_attw<<<g3, 256, 0, stream>>>(Qw, Kw, Mw, Lw, att);

  dim3 g4(Dn / 64, (Bn * Sn) / 128);
  mhsa_oproj<<<g4, 256, 0, stream>>>(OH, ow_w, ow_b, out);
}